// DirectModel_58909771432127
// MI455X (gfx1250) — compile-verified
//
#include <hip/hip_runtime.h>

// ---------------------------------------------------------------------------
// Template-MSE argmin, MI455X (gfx1250).
// argmin_t ||x-t||^2 = argmin_t (||t||^2 - 2 x.t)  (||x||^2 drops out).
// cross = x.t^T via V_WMMA_F32_16X16X4_F32 (exact fp32).
// Memory-bound: 822 MB of x @ 23.3 TB/s ~= 35 us floor.  x is streamed
// HBM -> LDS with GLOBAL_LOAD_ASYNC_TO_LDS_B128 (ASYNCcnt), double-buffered
// per wave so global latency hides behind the previous chunk's WMMAs.
// ---------------------------------------------------------------------------

typedef float v8f __attribute__((ext_vector_type(8)));
typedef float v2f __attribute__((ext_vector_type(2)));

#if defined(__AMDGCN__) && __has_builtin(__builtin_amdgcn_wmma_f32_16x16x4_f32)
#define HAVE_WMMA_F32X4 1
#else
#define HAVE_WMMA_F32X4 0
#endif

#define NUM_T   10
#define PIX     784
#define TSTRIDE 788      // 788 % 64 == 20 == 4*5  (odd*4) -> conflict-free b64 frags
#define KCHUNK  56       // 784 = 14 * 56 ; 14 wmma (K=4) steps per chunk
#define NCHUNK  14
#define XSTRIDE 60       // 60 % 64 == 60 == 4*15 (odd*4) -> conflict-free
#define LOADS_PER_CHUNK 7    // 16 rows * 14 float4 = 224 float4 / 32 lanes
#define WAVES   8
#define TILES_PER_BLOCK 8

// dynamic LDS (floats):
//   t_lds 16*788=12608 | tsq 16 | psum 256 | xbuf 8*2*16*60=15360 | sbuf 8*256=2048
//   total 30288 floats = 121152 B  -> 2 blocks resident per 320KB WGP
#define XBUF_WAVE   (2 * 16 * XSTRIDE)
#define LDS_FLOATS  (16*TSTRIDE + 16 + 256 + WAVES*XBUF_WAVE + WAVES*256)

extern "C" __global__ void __launch_bounds__(256, 2)
mse_argmin_wmma(const float* __restrict__ x,
                const float* __restrict__ tmpl,
                float* __restrict__ out)
{
    extern __shared__ float lds[];
    float* t_lds  = lds;                           // [16][TSTRIDE] zero-padded
    float* tsq    = t_lds + 16 * TSTRIDE;          // [16]
    float* psum   = tsq + 16;                      // [16][16]
    float* xbuf_a = psum + 256;                    // [WAVES][2][16][XSTRIDE]
    float* sbuf_a = xbuf_a + WAVES * XBUF_WAVE;    // [WAVES][16][16]

    const int tid  = threadIdx.x;
    const int lane = tid & 31;
    const int w    = tid >> 5;

    // ---- zero-fill template pad region ----
    for (int i = tid; i < 16 * TSTRIDE; i += 256) t_lds[i] = 0.0f;
    __syncthreads();

    // ---- load templates, coalesced ----
    for (int i = tid; i < NUM_T * PIX; i += 256) {
        int r = i / PIX;
        int c = i - r * PIX;
        t_lds[r * TSTRIDE + c] = tmpl[i];
    }
    __syncthreads();

    // ---- ||t||^2 per template: deterministic 16x16 partial tree ----
    {
        const int n = tid & 15, j = tid >> 4;      // 784 = 16 * 49
        float s = 0.0f;
        const int base = n * TSTRIDE + j * 49;
        #pragma unroll 7
        for (int k = 0; k < 49; ++k) { float v = t_lds[base + k]; s += v * v; }
        psum[n * 16 + j] = s;
    }
    __syncthreads();
    if (tid < 16) {
        float s = 0.0f;
        #pragma unroll
        for (int j = 0; j < 16; ++j) s += psum[tid * 16 + j];
        tsq[tid] = s;
    }
    __syncthreads();

    // ---- main GEMM: one 16-sample tile per wave, double-buffered async DMA ----
    const int  tile = blockIdx.x * TILES_PER_BLOCK + w;
    const long row0 = (long)tile * 16;
    float* xwave = xbuf_a + w * XBUF_WAVE;

    const int nn   = lane & 15;          // A row M / B col N owned by this lane
    const int koff = (lane >> 4) << 1;   // 0 (lanes 0-15) or 2 (lanes 16-31)

    v8f acc = {0.f, 0.f, 0.f, 0.f, 0.f, 0.f, 0.f, 0.f};

    // issue one chunk's 7 per-lane async b128 copies HBM -> wave-private LDS
    auto stage = [&](int chunk, int p) {
        const int k0 = chunk * KCHUNK;
        float* dst = xwave + p * (16 * XSTRIDE);
        #pragma unroll
        for (int i = 0; i < LOADS_PER_CHUNK; ++i) {
            const int idx = i * 32 + lane;          // 0..223
            const int r   = idx / 14;               // 14 float4 per row
            const int c   = idx - r * 14;
            const unsigned long long ga =
                (unsigned long long)(const void*)(x + (row0 + r) * PIX + k0 + c * 4);
            const unsigned int la =
                (unsigned int)(unsigned long long)(const void*)(dst + r * XSTRIDE + c * 4);
            asm volatile("global_load_async_to_lds_b128 %0, %1, off"
                         :: "v"(la), "v"(ga) : "memory");
        }
    };

    // 14 x V_WMMA_F32_16X16X4_F32 over one staged chunk
    auto compute = [&](int chunk) {
        const float* cur = xwave + (chunk & 1) * (16 * XSTRIDE);
        const int k0 = chunk * KCHUNK;
        #pragma unroll
        for (int s = 0; s < 14; ++s) {
            const int k = s * 4 + koff;
            v2f a = *(const v2f*)(cur   + nn * XSTRIDE + k);       // A 16x4 frag
            v2f b = *(const v2f*)(t_lds + nn * TSTRIDE + k0 + k);  // B 4x16 frag
#if HAVE_WMMA_F32X4
            acc = __builtin_amdgcn_wmma_f32_16x16x4_f32(
                false, a, false, b, (short)0, acc, false, false);
#else
            acc[0] += a.x * b.x + a.y * b.y;   // host-pass placeholder
#endif
        }
    };

    stage(0, 0);
    for (int c = 0; c < NCHUNK - 1; ++c) {
        stage(c + 1, (c + 1) & 1);
        // async loads complete in order: <=7 outstanding => chunk c has landed
        asm volatile("s_wait_asynccnt 7" ::: "memory");
        compute(c);
    }
    asm volatile("s_wait_asynccnt 0" ::: "memory");
    compute(NCHUNK - 1);

    // ---- epilogue: score = ||t||^2 - 2*cross ; first-occurrence argmin ----
    float* sbuf = sbuf_a + w * 256;
    const int half = lane >> 4;
    #pragma unroll
    for (int v = 0; v < 8; ++v) {
        const int m = v + 8 * half;               // D layout: VGPR v -> M=v / M=v+8
        const float score = (nn < NUM_T) ? (tsq[nn] - 2.0f * acc[v]) : 3.0e38f;
        sbuf[m * 16 + nn] = score;
    }
    __syncthreads();

    if (lane < 16) {
        const int m = lane;
        float best = sbuf[m * 16];
        int   bi   = 0;
        #pragma unroll
        for (int n = 1; n < NUM_T; ++n) {
            const float sc = sbuf[m * 16 + n];
            if (sc < best) { best = sc; bi = n; } // strict < => first-min tie-break
        }
        out[row0 + m] = (float)bi;
    }
}

extern "C" void kernel_launch(void* const* d_in, const int* in_sizes, int n_in,
                              void* d_out, int out_size, void* d_ws, size_t ws_size,
                              hipStream_t stream) {
    (void)n_in; (void)d_ws; (void)ws_size; (void)out_size;
    const float* x = (const float*)d_in[0];
    const float* t = (const float*)d_in[1];
    float* out = (float*)d_out;

    const int B      = in_sizes[0] / PIX;            // 262144
    const int tiles  = B / 16;                       // 16384
    const int blocks = tiles / TILES_PER_BLOCK;      // 2048
    const size_t shmem = (size_t)LDS_FLOATS * sizeof(float);

    mse_argmin_wmma<<<blocks, 256, shmem, stream>>>(x, t, out);
}